// GrammarEncoder_9672266351355
// MI455X (gfx1250) — compile-verified
//
#include <hip/hip_runtime.h>
#include <hip/hip_bf16.h>

// ---------------------------------------------------------------------------
// GrammarEncoder for MI455X (gfx1250, wave32)
//   Stage 1: type-bucketed expert MLP via V_WMMA_F32_16X16X4_F32
//   Stage 2: GI = x_seq @ Wih^T + bih  (big fp32 WMMA GEMM, per layer,
//            A-tiles staged with GLOBAL_LOAD_ASYNC_TO_LDS_B128)
//   Stage 3: persistent-kernel GRU recurrence (64 blocks, grid barrier per
//            step; Whh lives in the 192MB L2)
//   Stage 4: final FC matvec
// Workspace: ~145 MB (perm + emb 16MB + GI 96MB reused + out0 32MB)
// ---------------------------------------------------------------------------

typedef __attribute__((ext_vector_type(2))) float v2f;
typedef __attribute__((ext_vector_type(8))) float v8f;
typedef __attribute__((ext_vector_type(4))) int   ge_v4i;

#define N_NODES 8192
#define IN_DIM  64
#define MLP_HID 128
#define EMB     512
#define HID     1024
#define GATES   3072   // 3*HID
#define N_TYPES 16
#define NB_GRU  64     // persistent blocks for the recurrence (16 lanes / j)

#if __has_builtin(__builtin_amdgcn_global_load_async_to_lds_b128) && \
    __has_builtin(__builtin_amdgcn_s_wait_asynccnt)
#define GE_ASYNC_LDS 1
typedef __attribute__((address_space(1))) ge_v4i* ge_gptr;   // global v4i*
typedef __attribute__((address_space(3))) ge_v4i* ge_lptr;   // LDS v4i*
#endif

// fp32 WMMA 16x16x4 helper: D = A(16x4) * B(4x16) + C
__device__ __forceinline__ v8f wmma_f32x4(v2f a, v2f b, v8f c) {
    return __builtin_amdgcn_wmma_f32_16x16x4_f32(
        /*neg_a=*/false, a, /*neg_b=*/false, b,
        /*c_mod=*/(short)0, c, /*reuse_a=*/false, /*reuse_b=*/false);
}

// ----------------------------- bucketing -----------------------------------
__global__ void ge_init_meta(int* meta) {
    if (threadIdx.x < 128) meta[threadIdx.x] = 0;
}

__global__ void ge_hist(const int* __restrict__ types, int* __restrict__ counts) {
    int i = blockIdx.x * blockDim.x + threadIdx.x;
    if (i < N_NODES) atomicAdd(&counts[types[i]], 1);
}

__global__ void ge_scan(const int* __restrict__ counts,
                        int* __restrict__ offsets, int* __restrict__ cursors) {
    if (threadIdx.x == 0) {
        int acc = 0;
        for (int t = 0; t < N_TYPES; ++t) {
            offsets[t] = acc; cursors[t] = acc; acc += counts[t];
        }
        offsets[N_TYPES] = acc;
    }
}

__global__ void ge_scatter(const int* __restrict__ types,
                           int* __restrict__ cursors, int* __restrict__ perm) {
    int i = blockIdx.x * blockDim.x + threadIdx.x;
    if (i < N_NODES) {
        int p = atomicAdd(&cursors[types[i]], 1);
        perm[p] = i;
    }
}

// --------------------- expert MLP (per-type WMMA GEMM) ----------------------
// grid = (512 tiles, 16 types), block = 256 (8 waves). Each block: 16 rows of
// one type bucket. Stage1: H = relu(A@W1[t]+b1)  (8 waves x 16 cols = 128)
// Stage2: EMB = H@W2[t]+b2 (8 waves x 4 n-tiles = 512 cols)
__global__ __launch_bounds__(256)
void ge_mlp(const float* __restrict__ feats,
            const float* __restrict__ W1, const float* __restrict__ b1,
            const float* __restrict__ W2, const float* __restrict__ b2,
            const int* __restrict__ perm, const int* __restrict__ offsets,
            float* __restrict__ emb) {
    const int t    = blockIdx.y;
    const int beg  = offsets[t], end = offsets[t + 1];
    const int tile0 = beg + blockIdx.x * 16;
    if (tile0 >= end) return;                 // uniform per block: EXEC stays full

    const int tid  = threadIdx.x;
    const int wave = tid >> 5, lane = tid & 31;
    const int mrow = lane & 15;               // row / N index for this lane
    const int kb   = (lane >> 4) * 2;         // K sub-base inside a 4-chunk
    const int moff = (lane >> 4) * 8;         // C/D row offset

    __shared__ float As[16][IN_DIM];
    __shared__ float Hs[16][MLP_HID + 4];
    __shared__ int   rows[16];

    if (tid < 16) {
        int idx = tile0 + tid;
        rows[tid] = (idx < end) ? perm[idx] : perm[tile0];  // duplicate tail
    }
    __syncthreads();
    for (int i = tid; i < 16 * IN_DIM; i += 256) {
        int r = i >> 6, c = i & 63;
        As[r][c] = feats[(size_t)rows[r] * IN_DIM + c];
    }
    __syncthreads();

    // Stage 1: H[16][128]
    {
        const int n0 = wave * 16;
        const float* Bt = W1 + (size_t)t * IN_DIM * MLP_HID;
        v8f acc = {};
        for (int k = 0; k < IN_DIM; k += 4) {
            v2f a, b;
            a[0] = As[mrow][k + kb];
            a[1] = As[mrow][k + kb + 1];
            const float* bp = Bt + (size_t)(k + kb) * MLP_HID + (n0 + mrow);
            b[0] = bp[0];
            b[1] = bp[MLP_HID];
            acc = wmma_f32x4(a, b, acc);
        }
        float bias = b1[t * MLP_HID + n0 + mrow];
#pragma unroll
        for (int v = 0; v < 8; ++v) {
            float val = acc[v] + bias;
            Hs[v + moff][n0 + mrow] = val > 0.f ? val : 0.f;
        }
    }
    __syncthreads();

    // Stage 2: EMB[16][512]
    {
        const float* Bt = W2 + (size_t)t * MLP_HID * EMB;
        for (int nt = 0; nt < 4; ++nt) {
            const int n0 = (wave * 4 + nt) * 16;
            v8f acc = {};
            for (int k = 0; k < MLP_HID; k += 4) {
                v2f a, b;
                a[0] = Hs[mrow][k + kb];
                a[1] = Hs[mrow][k + kb + 1];
                const float* bp = Bt + (size_t)(k + kb) * EMB + (n0 + mrow);
                b[0] = bp[0];
                b[1] = bp[EMB];
                acc = wmma_f32x4(a, b, acc);
            }
            float bias = b2[t * EMB + n0 + mrow];
#pragma unroll
            for (int v = 0; v < 8; ++v) {
                int m = v + moff;
                if (tile0 + m < end)
                    emb[(size_t)rows[m] * EMB + n0 + mrow] = acc[v] + bias;
            }
        }
    }
}

// ------------------- GI = X @ W^T + bias  (WMMA GEMM) -----------------------
// X: [8192, K] row-major; W: [3072, K] row-major; GI: [8192, 3072]
// grid = (512 node-tiles, 24 gate-groups of 128), block = 256 (8 waves x 16 n)
__global__ __launch_bounds__(256)
void ge_gi_gemm(const float* __restrict__ X, const float* __restrict__ W,
                const float* __restrict__ bias, float* __restrict__ GI, int K) {
    const int m0  = blockIdx.x * 16;
    const int tid = threadIdx.x, wave = tid >> 5, lane = tid & 31;
    const int mrow = lane & 15;
    const int kb   = (lane >> 4) * 2;
    const int moff = (lane >> 4) * 8;
    const int n0   = blockIdx.y * 128 + wave * 16;

    __shared__ float As[16][64];

    v8f acc = {};
    const float* wp0 = W + (size_t)(n0 + mrow) * K;
    for (int kc = 0; kc < K; kc += 64) {
        __syncthreads();
#ifdef GE_ASYNC_LDS
        {   // 16 rows x 64 floats = 4KB; one async b128 per lane
            int r = tid >> 4, c4 = (tid & 15) * 4;
            const float* src = X + (size_t)(m0 + r) * K + kc + c4;
            __builtin_amdgcn_global_load_async_to_lds_b128(
                (ge_gptr)src, (ge_lptr)&As[r][c4], 0, 0);
            __builtin_amdgcn_s_wait_asynccnt(0);
        }
#else
        for (int i = tid; i < 16 * 64; i += 256) {
            int r = i >> 6, c = i & 63;
            As[r][c] = X[(size_t)(m0 + r) * K + kc + c];
        }
#endif
        __syncthreads();
        const float* wp = wp0 + kc + kb;
#pragma unroll
        for (int k = 0; k < 64; k += 4) {
            v2f a, b;
            a[0] = As[mrow][k + kb];
            a[1] = As[mrow][k + kb + 1];
            b[0] = wp[k];
            b[1] = wp[k + 1];
            acc = wmma_f32x4(a, b, acc);
        }
    }
    float bv = bias[n0 + mrow];
#pragma unroll
    for (int v = 0; v < 8; ++v)
        GI[(size_t)(m0 + v + moff) * GATES + n0 + mrow] = acc[v] + bv;
}

// -------------------------- GRU recurrence ---------------------------------
__global__ void ge_gru_prep(float* __restrict__ hbuf,
                            const float* __restrict__ h0, int* __restrict__ bar) {
    int j = threadIdx.x;                 // 1024 threads
    hbuf[j] = h0[j];
    if (j == 0) *bar = 0;
}

// Persistent kernel: 64 blocks x 256 = 16384 lanes; 16 lanes per output j.
// Each group computes the 3 recurrent dot products (r,z,n rows of Whh),
// reduces with wave32 shuffles, applies gate math, then device-wide barrier.
__global__ __launch_bounds__(256)
void ge_gru_layer(const float* __restrict__ GI, const float* __restrict__ Whh,
                  const float* __restrict__ bhh, float* __restrict__ hbuf,
                  float* __restrict__ out_seq, int T, int* __restrict__ bar) {
    const int g   = blockIdx.x * 256 + threadIdx.x;
    const int j   = g >> 4;
    const int sub = g & 15;
    const int k0  = sub * 64;
    const float* w0 = Whh + (size_t)j * HID + k0;
    const float* w1 = Whh + (size_t)(HID + j) * HID + k0;
    const float* w2 = Whh + (size_t)(2 * HID + j) * HID + k0;
    const float b0 = bhh[j], b1v = bhh[HID + j], b2v = bhh[2 * HID + j];

    int barTarget = 0;
    for (int t = 0; t < T; ++t) {
        const float* h  = hbuf + (t & 1) * HID;
        float*       hn = hbuf + ((t + 1) & 1) * HID;
        const float* hk = h + k0;
        float a0 = 0.f, a1 = 0.f, a2 = 0.f;
#pragma unroll 8
        for (int k = 0; k < 64; ++k) {
            float hv = hk[k];
            a0 = fmaf(w0[k], hv, a0);
            a1 = fmaf(w1[k], hv, a1);
            a2 = fmaf(w2[k], hv, a2);
        }
#pragma unroll
        for (int s = 8; s; s >>= 1) {     // reduce the 16-lane subgroup
            a0 += __shfl_xor(a0, s, 32);
            a1 += __shfl_xor(a1, s, 32);
            a2 += __shfl_xor(a2, s, 32);
        }
        if (sub == 0) {
            const float* gi = GI + (size_t)t * GATES;
            __builtin_prefetch(GI + (size_t)(t + 1) * GATES + j, 0, 0);
            float r  = 1.f / (1.f + __expf(-(gi[j] + a0 + b0)));
            float z  = 1.f / (1.f + __expf(-(gi[HID + j] + a1 + b1v)));
            float n  = tanhf(gi[2 * HID + j] + r * (a2 + b2v));
            float hv = (1.f - z) * n + z * h[j];
            hn[j] = hv;
            if (out_seq) out_seq[(size_t)t * HID + j] = hv;
        }
        // device-wide barrier (monotonic counter; NB_GRU co-resident blocks)
        __threadfence();
        __syncthreads();
        barTarget += NB_GRU;
        if (threadIdx.x == 0) {
            atomicAdd(bar, 1);
            while (__hip_atomic_load(bar, __ATOMIC_ACQUIRE,
                                     __HIP_MEMORY_SCOPE_AGENT) < barTarget) {}
        }
        __syncthreads();
    }
}

// ------------------------------ final FC -----------------------------------
__global__ void ge_fc(const float* __restrict__ h, const float* __restrict__ W,
                      const float* __restrict__ b, float* __restrict__ out) {
    int i = blockIdx.x * blockDim.x + threadIdx.x;   // 0..1023
    const float* wr = W + (size_t)i * HID;
    float acc = 0.f;
    for (int k = 0; k < HID; ++k) acc = fmaf(wr[k], h[k], acc);
    out[i] = acc + b[i];
}

// ------------------------------- launch ------------------------------------
extern "C" void kernel_launch(void* const* d_in, const int* in_sizes, int n_in,
                              void* d_out, int out_size, void* d_ws, size_t ws_size,
                              hipStream_t stream) {
    const float* node_feats = (const float*)d_in[0];
    const int*   node_types = (const int*)  d_in[1];
    const float* W1   = (const float*)d_in[2];
    const float* b1   = (const float*)d_in[3];
    const float* W2   = (const float*)d_in[4];
    const float* b2   = (const float*)d_in[5];
    const float* Wih0 = (const float*)d_in[6];
    const float* Whh0 = (const float*)d_in[7];
    const float* bih0 = (const float*)d_in[8];
    const float* bhh0 = (const float*)d_in[9];
    const float* Wih1 = (const float*)d_in[10];
    const float* Whh1 = (const float*)d_in[11];
    const float* bih1 = (const float*)d_in[12];
    const float* bhh1 = (const float*)d_in[13];
    const float* h_init = (const float*)d_in[14];
    const float* fc_W = (const float*)d_in[15];
    const float* fc_b = (const float*)d_in[16];
    float* out = (float*)d_out;

    // workspace carve-up (~145 MB)
    char* ws = (char*)d_ws;
    int* meta    = (int*)ws;                    // [0:16) counts [16:32) cursors
    int* counts  = meta;                        // [32:49) offsets, [64] barrier
    int* cursors = meta + 16;
    int* offsets = meta + 32;
    int* bar     = meta + 64;
    int*   perm  = (int*)(ws + 4096);
    float* hbuf  = (float*)(ws + 4096 + N_NODES * 4);              // 2*HID
    float* emb   = hbuf + 2 * HID;                                 // 16 MB
    float* GI    = emb + (size_t)N_NODES * EMB;                    // 96 MB (reused)
    float* out0  = GI + (size_t)N_NODES * GATES;                   // 32 MB

    // 1) bucket nodes by type
    ge_init_meta<<<1, 128, 0, stream>>>(meta);
    ge_hist<<<N_NODES / 256, 256, 0, stream>>>(node_types, counts);
    ge_scan<<<1, 32, 0, stream>>>(counts, offsets, cursors);
    ge_scatter<<<N_NODES / 256, 256, 0, stream>>>(node_types, cursors, perm);

    // 2) expert MLP -> emb [8192,512]
    dim3 gmlp(N_NODES / 16, N_TYPES);
    ge_mlp<<<gmlp, 256, 0, stream>>>(node_feats, W1, b1, W2, b2, perm, offsets, emb);

    // 3) layer 0: GI0 = emb @ Wih0^T + bih0, then recurrence (stores out0 seq)
    dim3 ggi(N_NODES / 16, GATES / 128);
    ge_gi_gemm<<<ggi, 256, 0, stream>>>(emb, Wih0, bih0, GI, EMB);
    ge_gru_prep<<<1, HID, 0, stream>>>(hbuf, h_init, bar);
    ge_gru_layer<<<NB_GRU, 256, 0, stream>>>(GI, Whh0, bhh0, hbuf, out0, N_NODES, bar);

    // 4) layer 1: GI1 = out0 @ Wih1^T + bih1 (reuses GI buffer), recurrence
    ge_gi_gemm<<<ggi, 256, 0, stream>>>(out0, Wih1, bih1, GI, HID);
    ge_gru_prep<<<1, HID, 0, stream>>>(hbuf, h_init + HID, bar);
    ge_gru_layer<<<NB_GRU, 256, 0, stream>>>(GI, Whh1, bhh1, hbuf, nullptr, N_NODES, bar);

    // 5) FC on final h (T=8192 even -> final h in hbuf[0])
    ge_fc<<<HID / 256, 256, 0, stream>>>(hbuf, fc_W, fc_b, out);
}